// GTN_53609781789151
// MI455X (gfx1250) — compile-verified
//
#include <hip/hip_runtime.h>

// ---------------------------------------------------------------------------
// Problem constants (fixed by the reference's setup_inputs)
// ---------------------------------------------------------------------------
#define EN   5      // number of adjacency slices
#define CH   2      // channels
#define NV   3000   // total nodes (D + P)
#define DN   1000   // drugs
#define PN   2000   // proteins
#define FF   128    // feature width

typedef __attribute__((ext_vector_type(2))) float v2f;
typedef __attribute__((ext_vector_type(8))) float v8f;
typedef unsigned int u32x4 __attribute__((ext_vector_type(4)));
typedef int i32x8 __attribute__((ext_vector_type(8)));
typedef int i32x4 __attribute__((ext_vector_type(4)));

// ---------------------------------------------------------------------------
// fp32 WMMA wrapper: D = A(16x4) * B(4x16) + C(16x16), wave32
// ---------------------------------------------------------------------------
__device__ __forceinline__ v8f wmma_f32_k4(v2f a, v2f b, v8f c) {
  return __builtin_amdgcn_wmma_f32_16x16x4_f32(
      false, a, false, b, (short)0, c, false, false);
}

// ---------------------------------------------------------------------------
// Tensor Data Mover: async 2D tile load Global -> LDS (D# built in SGPRs).
//   tile0/tile1: tile extent (elements) in contiguous / row dimension
//   rem0/rem1:   remaining tensor extent from tile origin (OOB -> zero fill)
//   stride:      row stride in elements
//   pad_iv:      LDS padding interval code (4 => every 32 DWORDs, 5 => 64)
//                with pad_amount 0 => +1 DWORD, matching [BK+1]/[BN+1] pitch
// ---------------------------------------------------------------------------
__device__ __forceinline__ void tdm_load_2d(unsigned lds_off, const float* gptr,
                                            int rem0, int rem1,
                                            int tile0, int tile1,
                                            long stride_elems, int pad_iv) {
  unsigned long long ga = (unsigned long long)(uintptr_t)gptr;
  if (rem0 < 0) rem0 = 0;
  if (rem1 < 0) rem1 = 0;
  u32x4 g0;
  g0[0] = 1u;                                          // count=1, load, no gather
  g0[1] = lds_off;                                     // lds_addr (bytes)
  g0[2] = (unsigned)(ga & 0xFFFFFFFFu);                // global_addr[31:0]
  g0[3] = (unsigned)((ga >> 32) & 0x1FFFFFFu) | (2u << 30); // addr[56:32] | type=2
  i32x8 g1;
  g1[0] = (2 << 16) | (1 << 20) | (pad_iv << 22);      // data_size=4B, pad_en, amount=1dw
  g1[1] = (int)(((unsigned)rem0 & 0xFFFFu) << 16);     // tensor_dim0[15:0] @ bits63:48
  g1[2] = (int)((((unsigned)rem0 >> 16) & 0xFFFFu) |
                (((unsigned)rem1 & 0xFFFFu) << 16));   // dim0 hi | dim1 lo
  g1[3] = (int)((((unsigned)rem1 >> 16) & 0xFFFFu) |
                ((unsigned)tile0 << 16));              // dim1 hi | tile_dim0
  g1[4] = tile1;                                       // tile_dim1 | tile_dim2=0
  g1[5] = (int)((unsigned long long)stride_elems & 0xFFFFFFFFu); // dim0_stride lo
  g1[6] = (int)(((unsigned long long)stride_elems >> 32) & 0xFFFFu);
  g1[7] = 0;
  i32x4 z4 = {0, 0, 0, 0};
  i32x8 z8 = {0, 0, 0, 0, 0, 0, 0, 0};
  __builtin_amdgcn_tensor_load_to_lds(g0, g1, z4, z4, z8, 0);
}

// ---------------------------------------------------------------------------
// Generic tiled GEMM:  C = epilogue(A @ B [+ bias])
//   A: (M,K) row-major.  B: (K,N) row-major, or (N,K) row-major if TRANS_B.
//   Batched over blockIdx.z (element strides; stride 0 = broadcast).
//   EPI: 0 = none, 1 = bias + relu, 2 = sigmoid.
//   USE_TDM: stage tiles via tensor_load_to_lds, double-buffered (TENSORcnt).
// Block: 256 threads = 8 wave32 waves. Tile: 128x64, K-step 32.
// Wave w: rows [(w>>1)*32, +32), cols [(w&1)*32, +32) => 2x2 accumulators.
// ---------------------------------------------------------------------------
#define BM 128
#define BN 64
#define BK 32

template <int TRANS_B, int EPI, int USE_TDM>
__global__ __launch_bounds__(256)
void gemm_wmma_f32(const float* __restrict__ Ag, const float* __restrict__ Bg,
                   const float* __restrict__ bias, float* __restrict__ Cg,
                   int M, int N, int K,
                   long strideA, long strideB, long strideC) {
  Ag += (long)blockIdx.z * strideA;
  Bg += (long)blockIdx.z * strideB;
  Cg += (long)blockIdx.z * strideC;

  __shared__ float sA[2][BM][BK + 1];
  __shared__ float sB[2][BK][BN + 1];

  const int tid  = threadIdx.x;
  const int lane = tid & 31;
  const int wave = tid >> 5;            // 0..7
  const int wm   = (wave >> 1) * 32;    // 0,32,64,96
  const int wn   = (wave & 1) * 32;     // 0,32
  const int mrow = lane & 15;
  const int khalf = (lane < 16) ? 0 : 2;

  const int bm0 = blockIdx.x * BM;
  const int bn0 = blockIdx.y * BN;

  v8f acc00 = {}, acc01 = {}, acc10 = {}, acc11 = {};

  const int kTiles = (K + BK - 1) / BK;

  if (USE_TDM) {
    // prologue: DMA tile 0 into buffer 0 (wave 0 issues; TDM ignores EXEC)
    if (wave == 0) {
      tdm_load_2d((unsigned)(uintptr_t)&sA[0][0][0], Ag + (long)bm0 * K,
                  K, M - bm0, BK, BM, K, 4);
      tdm_load_2d((unsigned)(uintptr_t)&sB[0][0][0], Bg + bn0,
                  N - bn0, K, BN, BK, N, 5);
    }
  }

  for (int kt = 0; kt < kTiles; ++kt) {
    const int cur = USE_TDM ? (kt & 1) : 0;

    if (USE_TDM) {
      // issue DMA for the next K-tile into the other buffer, then wait for
      // the current tile's two loads (in-order TENSORcnt) and publish.
      if (wave == 0 && kt + 1 < kTiles) {
        const int nk = (kt + 1) * BK;
        tdm_load_2d((unsigned)(uintptr_t)&sA[cur ^ 1][0][0],
                    Ag + (long)bm0 * K + nk, K - nk, M - bm0, BK, BM, K, 4);
        tdm_load_2d((unsigned)(uintptr_t)&sB[cur ^ 1][0][0],
                    Bg + (long)nk * N + bn0, N - bn0, K - nk, BN, BK, N, 5);
      }
      if (kt + 1 < kTiles) __builtin_amdgcn_s_wait_tensorcnt((short)2);
      else                 __builtin_amdgcn_s_wait_tensorcnt((short)0);
      __syncthreads();
    } else {
      const int k0 = kt * BK;
      for (int idx = tid; idx < BM * BK; idx += 256) {
        int r = idx / BK, c = idx % BK;
        int gm = bm0 + r, gk = k0 + c;
        sA[0][r][c] = (gm < M && gk < K) ? Ag[(long)gm * K + gk] : 0.0f;
      }
      for (int idx = tid; idx < BK * BN; idx += 256) {
        int r = idx / BN, c = idx % BN;
        int gk = k0 + r, gn = bn0 + c;
        float v = 0.0f;
        if (gk < K && gn < N)
          v = TRANS_B ? Bg[(long)gn * K + gk] : Bg[(long)gk * N + gn];
        sB[0][r][c] = v;
      }
      if (kt + 1 < kTiles && tid < 64) {   // global_prefetch_b8 hint
        int nk = k0 + BK;
        int gm = bm0 + 2 * tid;
        if (gm < M && nk < K) __builtin_prefetch(&Ag[(long)gm * K + nk], 0, 1);
        if (TRANS_B) {
          int gn = bn0 + tid;
          if (gn < N && nk < K) __builtin_prefetch(&Bg[(long)gn * K + nk], 0, 1);
        }
      }
      __syncthreads();
    }

    // ---- 8 WMMA steps of K=4, 2x2 register tile per wave ----
#pragma unroll
    for (int kk = 0; kk < BK; kk += 4) {
      const int kb = kk + khalf;
      v2f a0, a1, b0, b1;
      a0.x = sA[cur][wm + mrow][kb];
      a0.y = sA[cur][wm + mrow][kb + 1];
      a1.x = sA[cur][wm + 16 + mrow][kb];
      a1.y = sA[cur][wm + 16 + mrow][kb + 1];
      b0.x = sB[cur][kb][wn + mrow];
      b0.y = sB[cur][kb + 1][wn + mrow];
      b1.x = sB[cur][kb][wn + 16 + mrow];
      b1.y = sB[cur][kb + 1][wn + 16 + mrow];
      acc00 = wmma_f32_k4(a0, b0, acc00);
      acc01 = wmma_f32_k4(a0, b1, acc01);
      acc10 = wmma_f32_k4(a1, b0, acc10);
      acc11 = wmma_f32_k4(a1, b1, acc11);
    }
    __syncthreads();
  }

  // ---- epilogue + store (C/D layout: VGPR r -> M = r, hi lanes M = r+8) ----
  const int col0 = bn0 + wn + mrow;
  const int col1 = col0 + 16;
  const int rb   = bm0 + wm + ((lane < 16) ? 0 : 8);
  const float bc0 = (EPI == 1 && col0 < N) ? bias[col0] : 0.0f;
  const float bc1 = (EPI == 1 && col1 < N) ? bias[col1] : 0.0f;
#pragma unroll
  for (int r = 0; r < 8; ++r) {
    int row0 = rb + r, row1 = rb + 16 + r;
    float v;
    if (row0 < M && col0 < N) {
      v = acc00[r];
      if (EPI == 1) v = fmaxf(v + bc0, 0.0f);
      if (EPI == 2) v = 1.0f / (1.0f + __expf(-v));
      Cg[(long)row0 * N + col0] = v;
    }
    if (row0 < M && col1 < N) {
      v = acc01[r];
      if (EPI == 1) v = fmaxf(v + bc1, 0.0f);
      if (EPI == 2) v = 1.0f / (1.0f + __expf(-v));
      Cg[(long)row0 * N + col1] = v;
    }
    if (row1 < M && col0 < N) {
      v = acc10[r];
      if (EPI == 1) v = fmaxf(v + bc0, 0.0f);
      if (EPI == 2) v = 1.0f / (1.0f + __expf(-v));
      Cg[(long)row1 * N + col0] = v;
    }
    if (row1 < M && col1 < N) {
      v = acc11[r];
      if (EPI == 1) v = fmaxf(v + bc1, 0.0f);
      if (EPI == 2) v = 1.0f / (1.0f + __expf(-v));
      Cg[(long)row1 * N + col1] = v;
    }
  }
}

// ---------------------------------------------------------------------------
// softmax over E for the three (C,E) weight tensors -> 30 floats
// ---------------------------------------------------------------------------
__global__ void gt_softmax_kernel(const float* __restrict__ w0a,
                                  const float* __restrict__ w0b,
                                  const float* __restrict__ w1,
                                  float* __restrict__ out) {
  if (threadIdx.x != 0 || blockIdx.x != 0) return;
  const float* srcs[3] = {w0a, w0b, w1};
  for (int s = 0; s < 3; ++s)
    for (int c = 0; c < CH; ++c) {
      float m = -1e30f;
      for (int e = 0; e < EN; ++e) m = fmaxf(m, srcs[s][c * EN + e]);
      float ex[EN], den = 0.0f;
      for (int e = 0; e < EN; ++e) { ex[e] = __expf(srcs[s][c * EN + e] - m); den += ex[e]; }
      float inv = 1.0f / den;
      for (int e = 0; e < EN; ++e) out[s * 10 + c * EN + e] = ex[e] * inv;
    }
}

// ---------------------------------------------------------------------------
// a0/b0/g1[c] = sum_e smw[...] * A[e]   (single pass over A)
// ---------------------------------------------------------------------------
__global__ void gt_combine_kernel(const float* __restrict__ A,
                                  const float* __restrict__ smw,
                                  float* __restrict__ a0,
                                  float* __restrict__ b0,
                                  float* __restrict__ g1) {
  const long nn = (long)NV * NV;
  long idx = (long)blockIdx.x * blockDim.x + threadIdx.x;
  if (idx >= nn) return;
  float av[EN];
#pragma unroll
  for (int e = 0; e < EN; ++e) av[e] = A[(long)e * nn + idx];
#pragma unroll
  for (int c = 0; c < CH; ++c) {
    float sa = 0, sb = 0, sg = 0;
#pragma unroll
    for (int e = 0; e < EN; ++e) {
      sa += smw[0 * 10 + c * EN + e] * av[e];
      sb += smw[1 * 10 + c * EN + e] * av[e];
      sg += smw[2 * 10 + c * EN + e] * av[e];
    }
    a0[(long)c * nn + idx] = sa;
    b0[(long)c * nn + idx] = sb;
    g1[(long)c * nn + idx] = sg;
  }
}

// ---------------------------------------------------------------------------
// per-row: H += I; row /= rowsum  (in place). grid = (N, C)
// ---------------------------------------------------------------------------
__global__ void norm_row_kernel(float* __restrict__ H, int n) {
  float* row = H + ((long)blockIdx.y * n + blockIdx.x) * n;
  const int i = blockIdx.x;
  __shared__ float red[256];
  float s = 0.0f;
  for (int j = threadIdx.x; j < n; j += 256)
    s += row[j] + ((j == i) ? 1.0f : 0.0f);
  red[threadIdx.x] = s;
  __syncthreads();
  for (int off = 128; off > 0; off >>= 1) {
    if (threadIdx.x < off) red[threadIdx.x] += red[threadIdx.x + off];
    __syncthreads();
  }
  float deg = red[0];
  if (deg <= 1e-10f) deg = 1.0f;
  const float inv = 1.0f / deg;
  for (int j = threadIdx.x; j < n; j += 256) {
    float v = row[j] + ((j == i) ? 1.0f : 0.0f);
    row[j] = v * inv;
  }
}

// ---------------------------------------------------------------------------
// SGC degree: dinv[i] = rsqrt(rowsum(Adj_b) + 2). grid = (N, B)
// ---------------------------------------------------------------------------
__global__ void sgc_dinv_kernel(const float* __restrict__ Adj, long stride,
                                float* __restrict__ dinv, int n) {
  const float* row = Adj + (long)blockIdx.y * stride + (long)blockIdx.x * n;
  __shared__ float red[256];
  float s = 0.0f;
  for (int j = threadIdx.x; j < n; j += 256) s += row[j];
  red[threadIdx.x] = s;
  __syncthreads();
  for (int off = 128; off > 0; off >>= 1) {
    if (threadIdx.x < off) red[threadIdx.x] += red[threadIdx.x + off];
    __syncthreads();
  }
  if (threadIdx.x == 0) {
    float deg = red[0] + 2.0f;
    if (deg <= 1e-10f) deg = 1.0f;
    dinv[(long)blockIdx.y * n + blockIdx.x] = rsqrtf(deg);
  }
}

// ---------------------------------------------------------------------------
// Ahat[i,j] = (Adj[i,j] + 2*(i==j)) * dinv[i] * dinv[j].  grid = (x, B)
// ---------------------------------------------------------------------------
__global__ void sgc_scale_kernel(const float* __restrict__ Adj, long strideAdj,
                                 const float* __restrict__ dinv,
                                 float* __restrict__ out, int n) {
  const long nn = (long)n * n;
  long idx = (long)blockIdx.x * blockDim.x + threadIdx.x;
  if (idx >= nn) return;
  const float* As = Adj + (long)blockIdx.y * strideAdj;
  const float* dv = dinv + (long)blockIdx.y * n;
  float* o = out + (long)blockIdx.y * nn;
  int i = (int)(idx / n);
  int j = (int)(idx % n);
  float v = As[idx] + ((i == j) ? 2.0f : 0.0f);
  o[idx] = v * dv[i] * dv[j];
}

// ---------------------------------------------------------------------------
// conv_sum = softmax(a_param)[0]*X1 + [1]*ch0 + [2]*ch1
// ---------------------------------------------------------------------------
__global__ void att_combine_kernel(const float* __restrict__ ap,
                                   const float* __restrict__ x1,
                                   const float* __restrict__ ch,
                                   float* __restrict__ out) {
  const long nf = (long)NV * FF;
  long idx = (long)blockIdx.x * blockDim.x + threadIdx.x;
  if (idx >= nf) return;
  float a0 = ap[0], a1 = ap[1], a2 = ap[2];
  float m = fmaxf(a0, fmaxf(a1, a2));
  float e0 = __expf(a0 - m), e1 = __expf(a1 - m), e2 = __expf(a2 - m);
  float inv = 1.0f / (e0 + e1 + e2);
  out[idx] = inv * (e0 * x1[idx] + e1 * ch[idx] + e2 * ch[nf + idx]);
}

// ---------------------------------------------------------------------------
// Host-side orchestration
// ---------------------------------------------------------------------------
static inline dim3 gemm_grid(int M, int N, int Z) {
  return dim3((M + BM - 1) / BM, (N + BN - 1) / BN, Z);
}

extern "C" void kernel_launch(void* const* d_in, const int* in_sizes, int n_in,
                              void* d_out, int out_size, void* d_ws, size_t ws_size,
                              hipStream_t stream) {
  (void)in_sizes; (void)n_in; (void)out_size; (void)ws_size;

  const float* A    = (const float*)d_in[0];   // (5, N, N)
  const float* DTI  = (const float*)d_in[1];   // (N, N)
  const float* dstr = (const float*)d_in[2];   // (D, F)
  const float* pstr = (const float*)d_in[3];   // (P, F)
  const float* w0a  = (const float*)d_in[4];   // (C, E)
  const float* w0b  = (const float*)d_in[5];
  const float* w1   = (const float*)d_in[6];
  const float* ap   = (const float*)d_in[7];   // (C+1)
  const float* wd1 = (const float*)d_in[10]; const float* bd1 = (const float*)d_in[11];
  const float* wd2 = (const float*)d_in[12]; const float* bd2 = (const float*)d_in[13];
  const float* wd3 = (const float*)d_in[14]; const float* bd3 = (const float*)d_in[15];
  const float* wp1 = (const float*)d_in[16]; const float* bp1 = (const float*)d_in[17];
  const float* wp2 = (const float*)d_in[18]; const float* bp2 = (const float*)d_in[19];
  const float* wp3 = (const float*)d_in[20]; const float* bp3 = (const float*)d_in[21];

  const long NN = (long)NV * NV;
  const long NF = (long)NV * FF;

  // workspace carving (fp32). ~299 MB total.
  float* ws   = (float*)d_ws;
  float* a0   = ws;            // 2*NN  (later reused for adj = Hn @ g1)
  float* b0   = a0 + 2 * NN;   // 2*NN  (later reused for per-channel Ahat)
  float* g1   = b0 + 2 * NN;   // 2*NN  (first NN later reused for Ahat_DTI)
  float* Hbuf = g1 + 2 * NN;   // 2*NN
  float* feat = Hbuf + 2 * NN; // NF
  float* t0   = feat + NF;     // NF  } t0,t1 contiguous -> batched stride NF
  float* t1   = t0 + NF;       // NF  }
  float* x1   = t1 + NF;       // NF
  float* chb  = x1 + NF;       // 2*NF (ch0, ch1 contiguous)
  float* cs   = chb + 2 * NF;  // NF
  float* dinv = cs + NF;       // 2*N
  float* smw  = dinv + 2 * NV; // 30

  const dim3 blk(256);
  const int nnBlocks = (int)((NN + 255) / 256);
  const int nfBlocks = (int)((NF + 255) / 256);

  // 1) softmax of GT weights, then single-pass weighted combine of A
  gt_softmax_kernel<<<1, 32, 0, stream>>>(w0a, w0b, w1, smw);
  gt_combine_kernel<<<nnBlocks, blk, 0, stream>>>(A, smw, a0, b0, g1);

  // 2) MLPs -> feature (relu((X @ W^T) + b), 3 layers each)
  gemm_wmma_f32<1, 1, 0><<<gemm_grid(DN, FF, 1), blk, 0, stream>>>(dstr, wd1, bd1, t0, DN, FF, FF, 0, 0, 0);
  gemm_wmma_f32<1, 1, 0><<<gemm_grid(DN, FF, 1), blk, 0, stream>>>(t0,   wd2, bd2, t1, DN, FF, FF, 0, 0, 0);
  gemm_wmma_f32<1, 1, 0><<<gemm_grid(DN, FF, 1), blk, 0, stream>>>(t1,   wd3, bd3, feat, DN, FF, FF, 0, 0, 0);
  gemm_wmma_f32<1, 1, 0><<<gemm_grid(PN, FF, 1), blk, 0, stream>>>(pstr, wp1, bp1, t0, PN, FF, FF, 0, 0, 0);
  gemm_wmma_f32<1, 1, 0><<<gemm_grid(PN, FF, 1), blk, 0, stream>>>(t0,   wp2, bp2, t1, PN, FF, FF, 0, 0, 0);
  gemm_wmma_f32<1, 1, 0><<<gemm_grid(PN, FF, 1), blk, 0, stream>>>(t1,   wp3, bp3, feat + (long)DN * FF, PN, FF, FF, 0, 0, 0);

  // 3) H = a0 @ b0  (dominant GEMM: fp32 WMMA + TDM double-buffered staging)
  gemm_wmma_f32<0, 0, 1><<<gemm_grid(NV, NV, CH), blk, 0, stream>>>(a0, b0, nullptr, Hbuf, NV, NV, NV, NN, NN, NN);

  // 4) Hn = row-normalize(H + I)   (in place)
  norm_row_kernel<<<dim3(NV, CH), blk, 0, stream>>>(Hbuf, NV);

  // 5) adj = Hn @ g1   -> reuse a0 buffer
  gemm_wmma_f32<0, 0, 1><<<gemm_grid(NV, NV, CH), blk, 0, stream>>>(Hbuf, g1, nullptr, a0, NV, NV, NV, NN, NN, NN);

  // 6) X_conv1 = sgc(sgc(feature, DTI), DTI)
  sgc_dinv_kernel<<<dim3(NV, 1), blk, 0, stream>>>(DTI, 0, dinv, NV);
  sgc_scale_kernel<<<dim3(nnBlocks, 1), blk, 0, stream>>>(DTI, 0, dinv, g1, NV);
  gemm_wmma_f32<0, 0, 1><<<gemm_grid(NV, FF, 1), blk, 0, stream>>>(g1, feat, nullptr, t0, NV, FF, NV, 0, 0, 0);
  gemm_wmma_f32<0, 0, 1><<<gemm_grid(NV, FF, 1), blk, 0, stream>>>(g1, t0,   nullptr, x1, NV, FF, NV, 0, 0, 0);

  // 7) chs[c] = sgc(sgc(feature, adj[c]), adj[c])   (batched over channels)
  sgc_dinv_kernel<<<dim3(NV, CH), blk, 0, stream>>>(a0, NN, dinv, NV);
  sgc_scale_kernel<<<dim3(nnBlocks, CH), blk, 0, stream>>>(a0, NN, dinv, b0, NV);
  gemm_wmma_f32<0, 0, 1><<<gemm_grid(NV, FF, CH), blk, 0, stream>>>(b0, feat, nullptr, t0, NV, FF, NV, NN, 0, NF);
  gemm_wmma_f32<0, 0, 1><<<gemm_grid(NV, FF, CH), blk, 0, stream>>>(b0, t0,   nullptr, chb, NV, FF, NV, NN, NF, NF);

  // 8) attention-weighted sum (C==2 => X_conv order is [X1, ch0, ch1])
  att_combine_kernel<<<nfBlocks, blk, 0, stream>>>(ap, x1, chb, cs);

  // 9) y = sigmoid(drug_feat @ prot_feat^T)
  gemm_wmma_f32<1, 2, 0><<<gemm_grid(DN, PN, 1), blk, 0, stream>>>(cs, cs + (long)DN * FF, nullptr, (float*)d_out, DN, PN, FF, 0, 0, 0);
}